// GNNModel_11493332484123
// MI455X (gfx1250) — compile-verified
//
#include <hip/hip_runtime.h>
#include <hip/hip_bf16.h>
#include <math.h>

typedef __attribute__((ext_vector_type(2))) float v2f;
typedef __attribute__((ext_vector_type(8))) float v8f;

// ---------------- degree kernels ----------------
__global__ void k_init_deg(float* __restrict__ deg, int n) {
    int i = blockIdx.x * blockDim.x + threadIdx.x;
    if (i < n) deg[i] = 1.0f;                 // self-loop counts as 1
}

__global__ void k_count_deg(const int* __restrict__ dst, float* __restrict__ deg, int E) {
    int e = blockIdx.x * blockDim.x + threadIdx.x;
    if (e < E) atomicAdd(&deg[dst[e]], 1.0f);
}

__global__ void k_rsqrt(float* __restrict__ deg, int n) {
    int i = blockIdx.x * blockDim.x + threadIdx.x;
    if (i < n) deg[i] = rsqrtf(deg[i]);
}

// ---------------- WMMA GEMM: H = X @ W (64x64), optional +bias, optional AGG = H*dis^2 ----
// One wave computes a 16(M) x 64(N) strip using V_WMMA_F32_16X16X4_F32.
// 4 waves / block -> 64 rows per block. W staged in LDS.
__global__ __launch_bounds__(128) void k_gemm64(
    const float* __restrict__ X, const float* __restrict__ W,
    const float* __restrict__ bias,          // may be null
    const float* __restrict__ dis,           // may be null
    float* __restrict__ H, float* __restrict__ AGG,  // AGG may be null
    int n_rows, int apply_bias)
{
    __shared__ float Wlds[64 * 64];
    for (int i = threadIdx.x; i < 64 * 64; i += 128) Wlds[i] = W[i];
    __syncthreads();

    const int wave = threadIdx.x >> 5;
    const int lane = threadIdx.x & 31;
    const int tile = blockIdx.x * 4 + wave;      // 16-row tile index (wave-uniform)
    const int row0 = tile * 16;
    if (row0 >= n_rows) return;                  // uniform branch: EXEC stays all-1 inside

    const int m    = lane & 15;                  // M (for A) / N (for B) within tile
    const int koff = (lane >> 4) * 2;            // K sub-offset per ISA A/B layout

    const float* __restrict__ xrow = X + (size_t)(row0 + m) * 64;

    v8f acc0 = {}, acc1 = {}, acc2 = {}, acc3 = {};
    #pragma unroll
    for (int kk = 0; kk < 64; kk += 4) {
        v2f a;
        a.x = xrow[kk + koff];
        a.y = xrow[kk + koff + 1];
        const float* wr0 = &Wlds[(kk + koff) * 64];
        const float* wr1 = &Wlds[(kk + koff + 1) * 64];
        v2f b0; b0.x = wr0[ 0 + m]; b0.y = wr1[ 0 + m];
        v2f b1; b1.x = wr0[16 + m]; b1.y = wr1[16 + m];
        v2f b2; b2.x = wr0[32 + m]; b2.y = wr1[32 + m];
        v2f b3; b3.x = wr0[48 + m]; b3.y = wr1[48 + m];
        acc0 = __builtin_amdgcn_wmma_f32_16x16x4_f32(false, a, false, b0, (short)0, acc0, false, false);
        acc1 = __builtin_amdgcn_wmma_f32_16x16x4_f32(false, a, false, b1, (short)0, acc1, false, false);
        acc2 = __builtin_amdgcn_wmma_f32_16x16x4_f32(false, a, false, b2, (short)0, acc2, false, false);
        acc3 = __builtin_amdgcn_wmma_f32_16x16x4_f32(false, a, false, b3, (short)0, acc3, false, false);
    }

    // C/D layout: element r at lane l -> row = r + 8*(l>>4), col = (l&15) + 16*nt
    const int rbase = (lane >> 4) * 8;
    #pragma unroll
    for (int r = 0; r < 8; ++r) {
        const int row = row0 + rbase + r;
        float d2 = 0.0f;
        if (dis) { float dv = dis[row]; d2 = dv * dv; }
        float* __restrict__ hrow = H + (size_t)row * 64;
        float* __restrict__ arow = AGG ? (AGG + (size_t)row * 64) : nullptr;
        float v0 = acc0[r], v1 = acc1[r], v2 = acc2[r], v3 = acc3[r];
        if (apply_bias) {
            v0 += bias[ 0 + m]; v1 += bias[16 + m];
            v2 += bias[32 + m]; v3 += bias[48 + m];
        }
        hrow[ 0 + m] = v0; hrow[16 + m] = v1; hrow[32 + m] = v2; hrow[48 + m] = v3;
        if (arow) {
            arow[ 0 + m] = v0 * d2; arow[16 + m] = v1 * d2;
            arow[32 + m] = v2 * d2; arow[48 + m] = v3 * d2;
        }
    }
}

// ---------------- edge scatter: AGG[dst] += H[src] * (dis[src]*dis[dst]) ----------------
// 16 threads per edge, 4 floats each (float4 gather + 4 atomic f32 adds, resolves in L2).
__global__ __launch_bounds__(256) void k_scatter(
    const float* __restrict__ H, float* __restrict__ AGG,
    const int* __restrict__ src, const int* __restrict__ dst,
    const float* __restrict__ dis, int n_edges)
{
    int t = blockIdx.x * blockDim.x + threadIdx.x;
    int e = t >> 4;
    if (e >= n_edges) return;
    int f = (t & 15) * 4;
    int s = src[e], d = dst[e];
    float norm = dis[s] * dis[d];
    const float4 hv = *(const float4*)(H + (size_t)s * 64 + f);
    float* ap = AGG + (size_t)d * 64 + f;
    atomicAdd(ap + 0, hv.x * norm);
    atomicAdd(ap + 1, hv.y * norm);
    atomicAdd(ap + 2, hv.z * norm);
    atomicAdd(ap + 3, hv.w * norm);
}

// ---------------- in-place bias + ReLU ----------------
__global__ void k_bias_relu(float* __restrict__ A, const float* __restrict__ b, int n_elems) {
    int t = blockIdx.x * blockDim.x + threadIdx.x;
    if (t >= n_elems) return;
    float v = A[t] + b[t & 63];
    A[t] = v > 0.0f ? v : 0.0f;
}

// ---------------- per-graph mean pool (block per graph, no atomics) ----------------
__global__ __launch_bounds__(256) void k_graph_mean(
    const float* __restrict__ H, const int* __restrict__ ptr,
    float* __restrict__ out, int n_graphs)
{
    int g = blockIdx.x;
    if (g >= n_graphs) return;
    int beg = ptr[g], end = ptr[g + 1];
    int col  = threadIdx.x & 63;
    int part = threadIdx.x >> 6;       // 0..3
    float s = 0.0f;
    for (int r = beg + part; r < end; r += 4)
        s += H[(size_t)r * 64 + col];
    __shared__ float red[256];
    red[threadIdx.x] = s;
    __syncthreads();
    if (part == 0) {
        float tot = red[col] + red[64 + col] + red[128 + col] + red[192 + col];
        out[g * 64 + col] = tot / (float)(end - beg);
    }
}

extern "C" void kernel_launch(void* const* d_in, const int* in_sizes, int n_in,
                              void* d_out, int out_size, void* d_ws, size_t ws_size,
                              hipStream_t stream) {
    const float* x   = (const float*)d_in[0];
    const int*   ei  = (const int*)  d_in[1];
    const int*   ptr = (const int*)  d_in[2];
    const float* W1  = (const float*)d_in[3];
    const float* b1  = (const float*)d_in[4];
    const float* W2  = (const float*)d_in[5];
    const float* b2  = (const float*)d_in[6];
    const float* Wf  = (const float*)d_in[7];
    const float* bf  = (const float*)d_in[8];
    float* out = (float*)d_out;

    const int N = in_sizes[0] / 64;       // 100000
    const int E = in_sizes[1] / 2;        // 3200000
    const int G = in_sizes[2] - 1;        // 125
    const int* src = ei;
    const int* dst = ei + E;

    float* ws  = (float*)d_ws;
    float* dis = ws;                        // [N]
    float* B0  = ws + N;                    // [N*64] : H
    float* B1  = B0 + (size_t)N * 64;       // [N*64] : AGG / layer input

    // degrees -> deg_inv_sqrt
    k_init_deg <<<(N + 255) / 256, 256, 0, stream>>>(dis, N);
    k_count_deg<<<(E + 255) / 256, 256, 0, stream>>>(dst, dis, E);
    k_rsqrt    <<<(N + 255) / 256, 256, 0, stream>>>(dis, N);

    const int gemm_blocks = (N / 16 + 3) / 4;
    const int sc_blocks   = (int)(((long long)E * 16 + 255) / 256);
    const int ew_blocks   = (N * 64 + 255) / 256;

    // layer 1
    k_gemm64  <<<gemm_blocks, 128, 0, stream>>>(x,  W1, nullptr, dis, B0, B1, N, 0);
    k_scatter <<<sc_blocks,   256, 0, stream>>>(B0, B1, src, dst, dis, E);
    k_bias_relu<<<ew_blocks,  256, 0, stream>>>(B1, b1, N * 64);

    // layer 2 (AGG written in-place over input: per-wave row ownership, no hazard)
    k_gemm64  <<<gemm_blocks, 128, 0, stream>>>(B1, W2, nullptr, dis, B0, B1, N, 0);
    k_scatter <<<sc_blocks,   256, 0, stream>>>(B0, B1, src, dst, dis, E);
    k_bias_relu<<<ew_blocks,  256, 0, stream>>>(B1, b2, N * 64);

    // final linear + per-graph mean
    k_gemm64  <<<gemm_blocks, 128, 0, stream>>>(B1, Wf, bf, nullptr, B0, nullptr, N, 1);
    k_graph_mean<<<G, 256, 0, stream>>>(B0, ptr, out, G);
}